// CentroidHead_81423989997657
// MI455X (gfx1250) — compile-verified
//
#include <hip/hip_runtime.h>
#include <hip/hip_bf16.h>
#include <math.h>

// ---------------- problem constants (from reference) ----------------
#define N_VOX   300000
#define N_CLU   60000
#define LATENT  64
#define CLU_DIM 128
#define VB 2
#define VX 128
#define VY 128
#define VZ 128
#define CB 2
#define CX 64
#define CY 64
#define CZ 64
#define VCELLS (VB*VX*VY*VZ)   // 4,194,304
#define CCELLS (CB*CX*CY*CZ)   //   524,288
#define SCORE_T 0.1f
#define CENT_T  0.2f
#define TOPK    128
#define NEGV    (-1.0e30f)
#define TAKEN   (-3.0e38f)

// output layout (floats), concatenated in reference return order
#define SCORES_OFF 0
#define COORD_OFF  (N_VOX)                    // 300000
#define FEAT_OFF   (COORD_OFF + TOPK*4)       // 300512
#define PSCORE_OFF (FEAT_OFF + TOPK*CLU_DIM)  // 316896
#define PVALID_OFF (PSCORE_OFF + TOPK)        // 317024

typedef float v2f __attribute__((ext_vector_type(2)));
typedef float v8f __attribute__((ext_vector_type(8)));

// ---------------- kernel 1: init dense grids + counter ----------------
__global__ void fill_kernel(float* __restrict__ vgrid, int* __restrict__ cgrid,
                            int* __restrict__ counter) {
    int i = blockIdx.x * blockDim.x + threadIdx.x;
    if (i < VCELLS) vgrid[i] = NEGV;
    if (i < CCELLS) cgrid[i] = -1;
    if (i == 0) *counter = 0;
}

// ---------------- kernel 2: scatter cluster ids into dense grid ----------------
__global__ void cscatter_kernel(const int* __restrict__ cc, int* __restrict__ cgrid) {
    int i = blockIdx.x * blockDim.x + threadIdx.x;
    if (i >= N_CLU) return;
    int b = cc[i*4+0], x = cc[i*4+1], y = cc[i*4+2], z = cc[i*4+3];
    int key = ((b * CX + x) * CY + y) * CZ + z;
    cgrid[key] = i;
}

// ---------------- kernel 3: WMMA matvec + sigmoid + scatter masked score ----------------
// One wave (32 lanes) computes a 16-voxel tile of voxel_feats[16,64] @ w[64].
// B replicates w across all 16 N-columns so every D column holds the row dots.
__global__ void score_wmma_kernel(const float* __restrict__ vf,
                                  const int*   __restrict__ vc,
                                  const float* __restrict__ w,
                                  const float* __restrict__ bptr,
                                  float* __restrict__ scores,
                                  float* __restrict__ vgrid) {
    const int lane = threadIdx.x & 31;
    const int wave = blockIdx.x * (blockDim.x >> 5) + (threadIdx.x >> 5);
    const int NT = N_VOX / 16;           // 18750 tiles, exact
    if (wave >= NT) return;              // wave-uniform: EXEC stays all-1s for WMMA

    const int base  = wave * 16;
    const int row   = base + (lane & 15);
    const int kHalf = (lane < 16) ? 0 : 2;   // A layout: K=0,1 lanes 0-15; K=2,3 lanes 16-31
    const float* arow = vf + (size_t)row * LATENT + kHalf;

    v8f c = {};
#pragma unroll
    for (int kk = 0; kk < 16; ++kk) {       // K = 64 in chunks of 4
        v2f a  = *(const v2f*)(arow + kk * 4);
        v2f bm = *(const v2f*)(w + kk * 4 + kHalf);   // w replicated over N columns
        c = __builtin_amdgcn_wmma_f32_16x16x4_f32(
                /*neg_a=*/false, a, /*neg_b=*/false, bm,
                /*c_mod=*/(short)0, c, /*reuse_a=*/false, /*reuse_b=*/false);
    }

    // Every column of D is identical: c[r]@lane0 = dot(row r), c[r]@lane16 = dot(row r+8)
    float dot = 0.0f;
#pragma unroll
    for (int r = 0; r < 8; ++r) {
        float lo = __shfl(c[r], 0, 32);
        float hi = __shfl(c[r], 16, 32);
        if (lane == r)     dot = lo;
        if (lane == r + 8) dot = hi;
    }

    if (lane < 16) {
        float sc = 1.0f / (1.0f + expf(-(dot + bptr[0])));
        scores[row] = sc;
        int b = vc[row*4+0], x = vc[row*4+1], y = vc[row*4+2], z = vc[row*4+3];
        int key = ((b * VX + x) * VY + y) * VZ + z;
        vgrid[key] = (sc > SCORE_T) ? sc : NEGV;
    }
}

// ---------------- kernel 4: local-max peak test + candidate compaction ----------------
__global__ void peaks_kernel(const float* __restrict__ scores,
                             const int*   __restrict__ vc,
                             const float* __restrict__ vgrid,
                             float* __restrict__ cval, int* __restrict__ cidx,
                             int* __restrict__ counter) {
    int i = blockIdx.x * blockDim.x + threadIdx.x;
    if (i >= N_VOX) return;
    float s = scores[i];
    if (!(s > CENT_T)) return;           // s>0.2 implies mask (s>0.1); masked == s
    int b = vc[i*4+0], x = vc[i*4+1], y = vc[i*4+2], z = vc[i*4+3];
    float h = NEGV;
#pragma unroll
    for (int dx = -1; dx <= 1; ++dx)
#pragma unroll
        for (int dy = -1; dy <= 1; ++dy)
#pragma unroll
            for (int dz = -1; dz <= 1; ++dz) {
                int nx = x + dx, ny = y + dy, nz = z + dz;
                if (nx < 0 || nx >= VX || ny < 0 || ny >= VY || nz < 0 || nz >= VZ) continue;
                int key = ((b * VX + nx) * VY + ny) * VZ + nz;
                float v = vgrid[key];
                h = fmaxf(h, v);
            }
    if (h == s) {                        // hmax == masked: local max (incl. self)
        int p = atomicAdd(counter, 1);
        cval[p] = s;
        cidx[p] = i;
    }
}

// ---------------- kernel 5: single-block deterministic top-128 ----------------
// Comparator (value desc, index asc) makes output independent of atomic append order.
__global__ void topk_kernel(float* __restrict__ cval, const int* __restrict__ cidx,
                            const int* __restrict__ counter,
                            float* __restrict__ tval, int* __restrict__ tidx) {
    __shared__ float sv[1024];
    __shared__ int   si[1024];
    __shared__ int   sj[1024];
    const int tid = threadIdx.x;
    int M = *counter;
    if (M > N_VOX) M = N_VOX;

    for (int round = 0; round < TOPK; ++round) {
        float bv = TAKEN; int bi = 0x7fffffff; int bj = 0;
        for (int j = tid; j < M; j += 1024) {
            float v = cval[j];
            int   i = cidx[j];
            if (v > bv || (v == bv && i < bi)) { bv = v; bi = i; bj = j; }
        }
        sv[tid] = bv; si[tid] = bi; sj[tid] = bj;
        __syncthreads();
        for (int off = 512; off > 0; off >>= 1) {
            if (tid < off) {
                float ov = sv[tid+off]; int oi = si[tid+off];
                if (ov > sv[tid] || (ov == sv[tid] && oi < si[tid])) {
                    sv[tid] = ov; si[tid] = oi; sj[tid] = sj[tid+off];
                }
            }
            __syncthreads();
        }
        if (tid == 0) {
            if (sv[0] > 0.5f * NEGV) {      // real candidate found
                tval[round] = sv[0];
                tidx[round] = si[0];
                cval[sj[0]] = TAKEN;        // exclude from later rounds
            } else {
                tval[round] = NEGV;         // slot invalid
                tidx[round] = 0;
            }
        }
        __syncthreads();
    }
}

// ---------------- kernel 6: per-peak outputs + deferred 3x3x3 avg pool ----------------
__global__ void out_kernel(const float* __restrict__ tval, const int* __restrict__ tidx,
                           const int*   __restrict__ vc,
                           const int*   __restrict__ cc,
                           const float* __restrict__ cf,
                           const int*   __restrict__ inv_map,
                           const int*   __restrict__ cgrid,
                           float* __restrict__ out) {
    const int k = blockIdx.x;     // peak slot 0..127
    const int c = threadIdx.x;    // channel 0..127
    float v = tval[k];
    bool valid = v > 0.5f * NEGV;
    int safe = valid ? tidx[k] : 0;

    if (c < 4) out[COORD_OFF + k*4 + c] = (float)vc[safe*4 + c];
    if (c == 0) {
        out[PSCORE_OFF + k] = valid ? v : 0.0f;
        out[PVALID_OFF + k] = valid ? 1.0f : 0.0f;
    }

    float feat = 0.0f;
    if (valid) {                  // block-uniform branch
        int cid = inv_map[safe];
        int b = cc[cid*4+0], x = cc[cid*4+1], y = cc[cid*4+2], z = cc[cid*4+3];
        float acc = 0.0f; int cnt = 0;
        for (int dx = -1; dx <= 1; ++dx)
            for (int dy = -1; dy <= 1; ++dy)
                for (int dz = -1; dz <= 1; ++dz) {
                    int nx = x + dx, ny = y + dy, nz = z + dz;
                    if (nx < 0 || nx >= CX || ny < 0 || ny >= CY || nz < 0 || nz >= CZ) continue;
                    int key = ((b * CX + nx) * CY + ny) * CZ + nz;
                    int nid = cgrid[key];
                    if (nid >= 0) { acc += cf[(size_t)nid * CLU_DIM + c]; ++cnt; }
                }
        feat = acc / (float)(cnt > 0 ? cnt : 1);   // center always active -> cnt>=1
    }
    out[FEAT_OFF + k*CLU_DIM + c] = feat;
}

// ---------------- host launch ----------------
extern "C" void kernel_launch(void* const* d_in, const int* in_sizes, int n_in,
                              void* d_out, int out_size, void* d_ws, size_t ws_size,
                              hipStream_t stream) {
    const float* vf      = (const float*)d_in[0];  // voxel_feats   [300000,64]
    const int*   vc      = (const int*)  d_in[1];  // voxel_coords  [300000,4]
    const float* cf      = (const float*)d_in[2];  // cluster_feats [60000,128]
    const int*   cc      = (const int*)  d_in[3];  // cluster_coords[60000,4]
    const int*   inv_map = (const int*)  d_in[4];  // inv_map       [300000]
    const float* conv_w  = (const float*)d_in[5];  // conv_w        [64,1]
    const float* conv_b  = (const float*)d_in[6];  // conv_b        [1]
    float* out = (float*)d_out;

    char* ws = (char*)d_ws;
    float* vgrid   = (float*)(ws);                                   // 16,777,216 B
    int*   cgrid   = (int*)  (ws + 16777216);                        //  2,097,152 B
    float* cval    = (float*)(ws + 16777216 + 2097152);              //  1,200,000 B
    int*   cidx    = (int*)  (ws + 16777216 + 2097152 + 1200000);    //  1,200,000 B
    int*   counter = (int*)  (ws + 16777216 + 2097152 + 2400000);    //         16 B
    float* tval    = (float*)(ws + 16777216 + 2097152 + 2400016);    //        512 B
    int*   tidx    = (int*)  (ws + 16777216 + 2097152 + 2400528);    //        512 B

    float* scores = out + SCORES_OFF;   // sigmoid scores live directly in d_out[0:300000]

    // 1) init dense grids + candidate counter
    fill_kernel<<<VCELLS / 256, 256, 0, stream>>>(vgrid, cgrid, counter);
    // 2) scatter cluster ids
    cscatter_kernel<<<(N_CLU + 255) / 256, 256, 0, stream>>>(cc, cgrid);
    // 3) WMMA score + scatter masked scores (1 tile of 16 voxels per wave, 8 waves/block)
    {
        int tiles = N_VOX / 16;                 // 18750
        int blocks = (tiles + 7) / 8;           // 2344
        score_wmma_kernel<<<blocks, 256, 0, stream>>>(vf, vc, conv_w, conv_b, scores, vgrid);
    }
    // 4) local-max peak test + compaction
    peaks_kernel<<<(N_VOX + 255) / 256, 256, 0, stream>>>(scores, vc, vgrid, cval, cidx, counter);
    // 5) deterministic top-128
    topk_kernel<<<1, 1024, 0, stream>>>(cval, cidx, counter, tval, tidx);
    // 6) per-peak outputs + deferred avg pool over cluster neighborhoods
    out_kernel<<<TOPK, CLU_DIM, 0, stream>>>(tval, tidx, vc, cc, cf, inv_map, cgrid, out);

    (void)in_sizes; (void)n_in; (void)out_size; (void)ws_size;
}